// SplineUnit_12893491823232
// MI455X (gfx1250) — compile-verified
//
#include <hip/hip_runtime.h>
#include <cstdint>

// ---------------------------------------------------------------------------
// SplineUnit elementwise kernel for MI455X (gfx1250).
//
// Roofline: 2^25 fp32 elements, 268 MB of HBM traffic -> 11.5 us floor at
// 23.3 TB/s. The transcendental chain (2 sigmoids + 2 Hermites + sin/cos,
// x3 steps) makes a naive version compute-bound by ~5-8x, so the kernel is
// built around cutting VALU work:
//   * sigmoids via gfx1250 hardware v_tanh_f32 on an odd quintic of x
//     (atanh(erf(y)) ~= y*(k + a3 y^2 + a5 y^4), k = 2/sqrt(pi));
//     index = fma(tanh, 2.5, 2.5) directly in [0,5]
//   * Hermite interpolation in monomial form; the 4 coefficients of the
//     selected interval fetched with ONE ds_load_b128 (bank-conflict-free:
//     4 distinct 16B lines, equal addresses broadcast), 3 FMAs to evaluate
//   * native v_sin_f32 / v_cos_f32; velo*cos + x*velo*sin = velo*fma(x,s,c)
//   * 128-bit loads/stores; input staged through LDS with the CDNA5 async
//     path (global_load_async_to_lds_b128 + s_wait_asynccnt), double-buffered
//     per-thread slots (ASYNCcnt completes in order -> no barriers needed)
//   * 32-bit index math only (nvec <= 2^23)
// No WMMA: this workload has no matmul structure; the CDNA5-specific paths
// exercised are the async global->LDS ops, ds_load_b128 and v_tanh_f32.
// ---------------------------------------------------------------------------

#define TPB        256
#define NUM_STEPS  3
#define STEP_LEN   0.33333f

// gfx1250 hardware tanh (TRANS op).
__device__ __forceinline__ float tanh_hw(float x) {
  float r;
  asm("v_tanh_f32 %0, %1" : "=v"(r) : "v"(x));
  return r;
}

// Horner cubic: c.x + t*(c.y + t*(c.z + t*c.w))
__device__ __forceinline__ float herm(float4 c, float t) {
  return fmaf(fmaf(fmaf(c.w, t, c.z), t, c.y), t, c.x);
}

// One element through NUM_STEPS Euler steps. coef[0..3] = angle intervals,
// coef[4..7] = velocity intervals (monomial Hermite coefficients).
__device__ __forceinline__ float advance(float v, const float4* __restrict__ coef) {
#pragma unroll
  for (int s = 0; s < NUM_STEPS; ++s) {
    const float x  = v;
    const float x2 = x * x;
    // 5*ngd(x) = 2.5*(1+tanh(g1)), g1 = atanh(erf(x/sqrt2))
    // 5*nerf(x) = 2.5*(1+tanh(g2)), g2 = atanh(erf(x))
    const float g1 = x * fmaf(fmaf(-3.7079e-5f, x2, 0.036335172f), x2, 0.79788456f);
    const float g2 = x * fmaf(fmaf(-2.0975e-4f, x2, 0.10277203f),  x2, 1.12837917f);
    const float i1 = fmaf(tanh_hw(g1), 2.5f, 2.5f);   // in [0,5]
    const float i2 = fmaf(tanh_hw(g2), 2.5f, 2.5f);   // in [0,5]
    // bgn = clamp(floor(i), 0, 3). floor(i) >= 0 by construction (i >= 0),
    // so only the upper clamp is needed; t = i - bgn (extrapolates like ref).
    const float f1 = fminf(floorf(i1), 3.0f);
    const float f2 = fminf(floorf(i2), 3.0f);
    const float4 ca = coef[(int)f1];        // ds_load_b128
    const float4 cv = coef[4 + (int)f2];    // ds_load_b128
    const float ang  = herm(ca, i1 - f1);
    const float velo = herm(cv, i2 - f2);
    const float sn = __sinf(ang);           // v_sin_f32
    const float cs = __cosf(ang);           // v_cos_f32
    v = fmaf(velo * fmaf(x, sn, cs), STEP_LEN, v);
  }
  return v;
}

__global__ __launch_bounds__(TPB) void spline_unit_kernel(
    const float* __restrict__ x,
    const float* __restrict__ angles,   const float* __restrict__ velocity,
    const float* __restrict__ dangles,  const float* __restrict__ dvelocity,
    float* __restrict__ out, int nvec /* number of float4 elements */) {
  __shared__ float4 s_coef[8];          // [0..3] angle, [4..7] velocity
  __shared__ float4 s_stage[2][TPB];    // double-buffered async staging

  const int tid = threadIdx.x;

  // Per-interval monomial Hermite coefficients, once per block:
  //   h(t) = p0 + m0 t + (3(p1-p0)-2m0-m1) t^2 + (2(p0-p1)+m0+m1) t^3
  if (tid < 8) {
    const float* v = (tid < 4) ? angles  : velocity;
    const float* d = (tid < 4) ? dangles : dvelocity;
    const int k = tid & 3;
    const float p0 = v[k], p1 = v[k + 1];
    const float m0 = d[k], m1 = d[k + 1];
    float4 c;
    c.x = p0;
    c.y = m0;
    c.z = fmaf(3.0f, p1 - p0, fmaf(-2.0f, m0, -m1));
    c.w = fmaf(2.0f, p0 - p1, m0 + m1);
    s_coef[tid] = c;
  }
  __syncthreads();

  const float4* __restrict__ xin  = (const float4*)x;
  float4* __restrict__       vout = (float4*)out;

  const int gstride = (int)(gridDim.x * TPB);        // fits in 32 bits
  const int i0      = (int)(blockIdx.x * TPB) + tid;
  if (i0 >= nvec) return;   // no barriers below, safe to retire early

  // LDS byte addresses of this thread's two staging slots: low 32 bits of the
  // flat shared-aperture pointer carry the LDS offset (ISA sec. 10.2).
  const uint32_t slot0 = (uint32_t)(uintptr_t)(&s_stage[0][tid]);
  const uint32_t slot1 = (uint32_t)(uintptr_t)(&s_stage[1][tid]);

  // Prologue: issue async load of tile 0 into slot 0.
  {
    const float4* ga = xin + i0;
    asm volatile("global_load_async_to_lds_b128 %0, %1, off"
                 :: "v"(slot0), "v"(ga) : "memory");
  }

  int stage = 0;
  for (int i = i0; i < nvec; i += gstride) {
    const int nxt = i + gstride;
    if (nxt < nvec) {
      // Fill the other buffer, then drain to 1 outstanding: ASYNCcnt retires
      // in order, so the current tile is guaranteed to be in LDS.
      const uint32_t lds_a = stage ? slot0 : slot1;
      const float4*  ga    = xin + nxt;
      asm volatile("global_load_async_to_lds_b128 %0, %1, off"
                   :: "v"(lds_a), "v"(ga) : "memory");
      asm volatile("s_wait_asynccnt 1" ::: "memory");
    } else {
      asm volatile("s_wait_asynccnt 0" ::: "memory");
    }

    float4 d = s_stage[stage][tid];       // ds_load_b128 from own slot
    d.x = advance(d.x, s_coef);
    d.y = advance(d.y, s_coef);
    d.z = advance(d.z, s_coef);
    d.w = advance(d.w, s_coef);
    vout[i] = d;                          // global_store_b128
    stage ^= 1;
  }
}

extern "C" void kernel_launch(void* const* d_in, const int* in_sizes, int n_in,
                              void* d_out, int out_size, void* d_ws, size_t ws_size,
                              hipStream_t stream) {
  const float* x    = (const float*)d_in[0];
  const float* ang  = (const float*)d_in[1];
  const float* vel  = (const float*)d_in[2];
  const float* dang = (const float*)d_in[3];
  const float* dvel = (const float*)d_in[4];
  float* out = (float*)d_out;

  const int n    = in_sizes[0];     // 8*4096*1024 = 2^25, divisible by 4
  const int nvec = n >> 2;          // float4 count

  int blocks = (nvec + TPB - 1) / TPB;
  if (blocks > 4096) blocks = 4096; // grid-stride; ~8 tiles/thread at full size

  spline_unit_kernel<<<blocks, TPB, 0, stream>>>(x, ang, vel, dang, dvel, out, nvec);
}